// NearestNeighborSampler_57157424775582
// MI455X (gfx1250) — compile-verified
//
#include <hip/hip_runtime.h>
#include <stdint.h>

// Problem constants (fixed by the reference)
#define B_ROWS 1024             // x rows
#define QSIZE  32768            // queue rows
#define FDIM   256              // feature dim
#define QOFF   (QSIZE - B_ROWS) // 31744: q[j] = queue[j+1024] for j<QOFF else x[j-QOFF]
#define LDS_STRIDE 260          // 256 + 4 pad, keeps 16B alignment (260*4 % 16 == 0)

typedef __attribute__((ext_vector_type(2))) float v2f;
typedef __attribute__((ext_vector_type(8))) float v8f;

// Monotone map float -> u32 so smaller float => smaller unsigned key (handles negatives).
__device__ __forceinline__ unsigned int fkey(float f) {
  unsigned int b = __float_as_uint(f);
  return (b & 0x80000000u) ? ~b : (b | 0x80000000u);
}

__device__ __forceinline__ const float* q_row(int j, const float* x, const float* queue) {
  return (j < QOFF) ? (queue + (size_t)(j + B_ROWS) * FDIM)
                    : (x + (size_t)(j - QOFF) * FDIM);
}

// ---------------------------------------------------------------------------
// Kernel 0: q2[j] = ||q_j||^2. One wave per row; 8 contiguous floats per lane.
// ---------------------------------------------------------------------------
__global__ __launch_bounds__(256) void nns_q2(const float* __restrict__ x,
                                              const float* __restrict__ queue,
                                              float* __restrict__ q2) {
  const int j    = blockIdx.x * 8 + (threadIdx.x >> 5);
  const int lane = threadIdx.x & 31;
  const float* src = q_row(j, x, queue);
  const float4 a = *(const float4*)(src + lane * 8);
  const float4 b = *(const float4*)(src + lane * 8 + 4);
  float s = a.x * a.x + a.y * a.y + a.z * a.z + a.w * a.w +
            b.x * b.x + b.y * b.y + b.z * b.z + b.w * b.w;
#pragma unroll
  for (int o = 1; o < 32; o <<= 1) s += __shfl_xor(s, o, 32);
  if (lane == 0) q2[j] = s;
}

// ---------------------------------------------------------------------------
// Kernel 1: init argmin slots.
// ---------------------------------------------------------------------------
__global__ void nns_init(unsigned long long* best) {
  int i = blockIdx.x * blockDim.x + threadIdx.x;
  if (i < B_ROWS) best[i] = 0xFFFFFFFFFFFFFFFFull;
}

// 8 WMMAs consuming one float4 fragment set (2 K-steps x 4 column tiles).
#define NNS_COMPUTE(aa, b0, b1, b2, b3)                                                        \
  do {                                                                                         \
    const v2f alo = {(aa).x, (aa).y};                                                          \
    const v2f ahi = {(aa).z, (aa).w};                                                          \
    c0 = __builtin_amdgcn_wmma_f32_16x16x4_f32(false, alo, false, (v2f){(b0).x, (b0).y}, (short)0, c0, false, false); \
    c1 = __builtin_amdgcn_wmma_f32_16x16x4_f32(false, alo, false, (v2f){(b1).x, (b1).y}, (short)0, c1, false, false); \
    c2 = __builtin_amdgcn_wmma_f32_16x16x4_f32(false, alo, false, (v2f){(b2).x, (b2).y}, (short)0, c2, false, false); \
    c3 = __builtin_amdgcn_wmma_f32_16x16x4_f32(false, alo, false, (v2f){(b3).x, (b3).y}, (short)0, c3, false, false); \
    c0 = __builtin_amdgcn_wmma_f32_16x16x4_f32(false, ahi, false, (v2f){(b0).z, (b0).w}, (short)0, c0, false, false); \
    c1 = __builtin_amdgcn_wmma_f32_16x16x4_f32(false, ahi, false, (v2f){(b1).z, (b1).w}, (short)0, c1, false, false); \
    c2 = __builtin_amdgcn_wmma_f32_16x16x4_f32(false, ahi, false, (v2f){(b2).z, (b2).w}, (short)0, c2, false, false); \
    c3 = __builtin_amdgcn_wmma_f32_16x16x4_f32(false, ahi, false, (v2f){(b3).z, (b3).w}, (short)0, c3, false, false); \
  } while (0)

// ---------------------------------------------------------------------------
// Kernel 2: WMMA score + argmin fold.
// Grid (QSIZE/512, B_ROWS/16), block = 256 threads = 8 waves; each wave owns
// four 16x16 f32 tiles (64 N-columns), K=256 via 64 x v_wmma_f32_16x16x4_f32.
//
// K-permutation: WMMA step t takes slot data (2t,2t+1) of the row's FIRST 128
// floats for lanes 0-15 and of the SECOND 128 floats for lanes 16-31 (A and B
// use the identical mapping, so the dot product over K is unchanged). Each
// lane therefore reads a contiguous stream -> b128 loads, 2 steps per load.
//
// Ping-pong double buffer + rolled loop (#pragma unroll 1): fragments are
// reloaded right after being consumed and stay live across the backedge, so
// every load is issued ~8 WMMAs before its use instead of immediately before.
// ---------------------------------------------------------------------------
__global__ __launch_bounds__(256) void nns_score(const float* __restrict__ x,
                                                 const float* __restrict__ queue,
                                                 const float* __restrict__ q2,
                                                 unsigned long long* __restrict__ best) {
  __shared__ float xs[16 * LDS_STRIDE];

  const int tid  = threadIdx.x;
  const int wave = tid >> 5;
  const int lane = tid & 31;
  const int mbase  = blockIdx.y * 16;
  const int nblock = blockIdx.x * 512;

  // Stage the 16x256 x-tile into LDS (row-major, padded stride).
  for (int i = tid; i < 16 * 64; i += 256) {
    const int row = i >> 6;
    const int c4  = (i & 63) << 2;
    const float4 v = *(const float4*)(x + (size_t)(mbase + row) * FDIM + c4);
    *(float4*)(&xs[row * LDS_STRIDE + c4]) = v;
  }
  __syncthreads();

  const int nwave = nblock + (wave << 6);      // 64 columns per wave
  const int coln  = nwave + (lane & 15);
  const int half  = (lane >> 4) << 7;          // 0 or 128: this lane's K-half

  const float* p0 = q_row(coln,      x, queue) + half;
  const float* p1 = q_row(coln + 16, x, queue) + half;
  const float* p2 = q_row(coln + 32, x, queue) + half;
  const float* p3 = q_row(coln + 48, x, queue) + half;
  const float* arow = &xs[(lane & 15) * LDS_STRIDE + half];

  v8f c0 = {}; v8f c1 = {}; v8f c2 = {}; v8f c3 = {};

  // Preload ping (tt=0) and pong (tt=1) fragment sets.
  float4 aA  = *(const float4*)(arow);
  float4 b0A = *(const float4*)(p0);
  float4 b1A = *(const float4*)(p1);
  float4 b2A = *(const float4*)(p2);
  float4 b3A = *(const float4*)(p3);

  float4 aB  = *(const float4*)(arow + 4);
  float4 b0B = *(const float4*)(p0 + 4);
  float4 b1B = *(const float4*)(p1 + 4);
  float4 b2B = *(const float4*)(p2 + 4);
  float4 b3B = *(const float4*)(p3 + 4);

#pragma unroll 1
  for (int i = 2; i <= 30; i += 2) {
    NNS_COMPUTE(aA, b0A, b1A, b2A, b3A);       // steps 2(i-2), 2(i-2)+1
    aA  = *(const float4*)(arow + 4 * i);      // prefetch set A <- tt=i
    b0A = *(const float4*)(p0 + 4 * i);
    b1A = *(const float4*)(p1 + 4 * i);
    b2A = *(const float4*)(p2 + 4 * i);
    b3A = *(const float4*)(p3 + 4 * i);
    NNS_COMPUTE(aB, b0B, b1B, b2B, b3B);       // steps 2(i-1), 2(i-1)+1
    aB  = *(const float4*)(arow + 4 * i + 4);  // prefetch set B <- tt=i+1
    b0B = *(const float4*)(p0 + 4 * i + 4);
    b1B = *(const float4*)(p1 + 4 * i + 4);
    b2B = *(const float4*)(p2 + 4 * i + 4);
    b3B = *(const float4*)(p3 + 4 * i + 4);
#if __has_builtin(__builtin_amdgcn_sched_group_barrier)
    // Pin the pipeline shape: WMMA x8, VMEM-read x4, DS-read x1, twice.
    __builtin_amdgcn_sched_group_barrier(8, 8, 0);
    __builtin_amdgcn_sched_group_barrier(32, 4, 0);
    __builtin_amdgcn_sched_group_barrier(256, 1, 0);
    __builtin_amdgcn_sched_group_barrier(8, 8, 0);
    __builtin_amdgcn_sched_group_barrier(32, 4, 0);
    __builtin_amdgcn_sched_group_barrier(256, 1, 0);
#endif
  }
  // Drain the two in-flight buffers (tt=30, tt=31).
  NNS_COMPUTE(aA, b0A, b1A, b2A, b3A);
  NNS_COMPUTE(aB, b0B, b1B, b2B, b3B);

  // Per-column squared norms (precomputed).
  const float n0 = q2[coln];
  const float n1 = q2[coln + 16];
  const float n2 = q2[coln + 32];
  const float n3 = q2[coln + 48];

  // C/D layout: VGPR v holds M = mbase+v (lanes 0-15) or mbase+v+8 (lanes 16-31), N = lane&15.
  const int rofs = (lane >> 4) << 3;           // 0 or 8
#pragma unroll
  for (int v = 0; v < 8; ++v) {
    const int row = mbase + v + rofs;
    float s0 = n0 - 2.0f * c0[v];              // q^2 - 2 x.q (x^2 constant per row)
    float s1 = n1 - 2.0f * c1[v];
    float s2 = n2 - 2.0f * c2[v];
    float s3 = n3 - 2.0f * c3[v];
    if (coln      == row + QOFF) s0 = __builtin_inff();   // mask self-match
    if (coln + 16 == row + QOFF) s1 = __builtin_inff();
    if (coln + 32 == row + QOFF) s2 = __builtin_inff();
    if (coln + 48 == row + QOFF) s3 = __builtin_inff();
    unsigned long long k0 = ((unsigned long long)fkey(s0) << 32) | (unsigned int)(coln);
    unsigned long long k1 = ((unsigned long long)fkey(s1) << 32) | (unsigned int)(coln + 16);
    unsigned long long k2 = ((unsigned long long)fkey(s2) << 32) | (unsigned int)(coln + 32);
    unsigned long long k3 = ((unsigned long long)fkey(s3) << 32) | (unsigned int)(coln + 48);
    unsigned long long m01 = (k1 < k0) ? k1 : k0;
    unsigned long long m23 = (k3 < k2) ? k3 : k2;
    unsigned long long bst = (m23 < m01) ? m23 : m01;
    // Butterfly min within each 16-lane half (the two halves cover different rows).
#pragma unroll
    for (int o = 1; o < 16; o <<= 1) {
      const unsigned long long other = __shfl_xor(bst, o, 32);
      if (other < bst) bst = other;
    }
    if ((lane & 15) == 0) atomicMin(&best[row], bst);
  }
}

// ---------------------------------------------------------------------------
// Kernel 3: one block per output row -> unpack winning column, copy the row.
// ---------------------------------------------------------------------------
__global__ void nns_gather(const float* __restrict__ x,
                           const float* __restrict__ queue,
                           const unsigned long long* __restrict__ best,
                           float* __restrict__ out) {
  const int row = blockIdx.x;
  const int idx = (int)(best[row] & 0xFFFFFFFFull);
  const float* src = q_row(idx, x, queue);
  const int t = threadIdx.x;                   // 64 threads * float4 = 256 floats
  const float4 v = *(const float4*)(src + t * 4);
  *(float4*)(out + (size_t)row * FDIM + t * 4) = v;
}

extern "C" void kernel_launch(void* const* d_in, const int* in_sizes, int n_in,
                              void* d_out, int out_size, void* d_ws, size_t ws_size,
                              hipStream_t stream) {
  const float* x     = (const float*)d_in[0];   // [1024, 256] f32
  const float* queue = (const float*)d_in[1];   // [32768, 256] f32
  float* out = (float*)d_out;                   // [1024, 256] f32

  // Workspace layout: [0, 128KB) = q2 (32768 f32); [128KB, 136KB) = argmin u64 slots.
  float* q2 = (float*)d_ws;
  unsigned long long* best =
      (unsigned long long*)((char*)d_ws + (size_t)QSIZE * sizeof(float));

  nns_q2<<<dim3(QSIZE / 8), dim3(256), 0, stream>>>(x, queue, q2);
  nns_init<<<dim3((B_ROWS + 255) / 256), dim3(256), 0, stream>>>(best);

  dim3 grid(QSIZE / 512, B_ROWS / 16);          // (64, 64) blocks, 8 waves each
  nns_score<<<grid, dim3(256), 0, stream>>>(x, queue, q2, best);

  nns_gather<<<dim3(B_ROWS), dim3(64), 0, stream>>>(x, queue, best, out);
}